// TurboDeepEPTokenDispatcher_73169062854666
// MI455X (gfx1250) — compile-verified
//
#include <hip/hip_runtime.h>
#include <stdint.h>

#define T_TOKENS  8192
#define H_DIM     2048
#define E_EXPERTS 64
#define K_TOP     8
#define OUT_ROWS  (T_TOKENS * K_TOP)   // 65536
#define ROW_BYTES (H_DIM * 4)          // 8192 bytes per row

// ---------------------------------------------------------------------------
// CDNA5 async global<->LDS copy path (ASYNCcnt). Guarded so compile never fails.
// ---------------------------------------------------------------------------
#if defined(__gfx1250__) && __has_builtin(__builtin_amdgcn_global_load_async_to_lds_b128) && \
    __has_builtin(__builtin_amdgcn_global_store_async_from_lds_b128)
#define USE_ASYNC_LDS 1
#else
#define USE_ASYNC_LDS 0
#endif

// CPol temporal hints (gfx12-family encoding: TH in [2:0]).
#define CPOL_RT 0   // regular temporal: keep hidden_states resident in L2 (8x reuse)
#define CPOL_NT 1   // non-temporal: 512 MiB write-once stream must not churn L2

typedef int v4i __attribute__((ext_vector_type(4)));
typedef __attribute__((address_space(1))) v4i* gptr_v4i;
typedef __attribute__((address_space(3))) v4i* lptr_v4i;

#define GPTR(p) ((gptr_v4i)(p))
#define LPTR(p) ((lptr_v4i)(p))

__device__ __forceinline__ void wait_asynccnt0() {
  asm volatile("s_wait_asynccnt 0" ::: "memory");
}

// ---------------------------------------------------------------------------
// Kernel 1: per-token top-8 of 64 probs -> 64-bit routing mask.
// Ties broken toward lower expert index (matches jax.lax.top_k stability).
// ---------------------------------------------------------------------------
__global__ void topk_mask_kernel(const float* __restrict__ token_probs,
                                 unsigned long long* __restrict__ masks) {
  int t = blockIdx.x * blockDim.x + threadIdx.x;
  if (t >= T_TOKENS) return;

  float p[E_EXPERTS];
  const float4* src = (const float4*)(token_probs + (size_t)t * E_EXPERTS);
#pragma unroll
  for (int i = 0; i < E_EXPERTS / 4; ++i) {
    float4 v = src[i];
    p[4 * i + 0] = v.x; p[4 * i + 1] = v.y;
    p[4 * i + 2] = v.z; p[4 * i + 3] = v.w;
  }

  unsigned long long mask = 0ull;
#pragma unroll 1
  for (int k = 0; k < K_TOP; ++k) {
    float best = -1.0f;   // probs are uniform [0,1)
    int   bi   = 0;
#pragma unroll
    for (int e = 0; e < E_EXPERTS; ++e) {
      if (p[e] > best) { best = p[e]; bi = e; }
    }
    mask |= (1ull << bi);
#pragma unroll
    for (int e = 0; e < E_EXPERTS; ++e) {
      if (e == bi) p[e] = -2.0f;   // remove winner, keep registers static-indexed
    }
  }
  masks[t] = mask;
}

// ---------------------------------------------------------------------------
// Kernel 2: per-expert counts + exclusive prefix (expert base offsets).
// 64 threads, one per expert; masks are L2-resident (64 KB).
// ---------------------------------------------------------------------------
__global__ void count_kernel(const unsigned long long* __restrict__ masks,
                             int* __restrict__ counts,
                             int* __restrict__ offsets,
                             float* __restrict__ out_counts) {
  int e = threadIdx.x;   // 0..63
  int c = 0;
  for (int t = 0; t < T_TOKENS; ++t) {
    c += (int)((masks[t] >> e) & 1ull);
  }
  counts[e] = c;
  out_counts[e] = (float)c;

  __shared__ int sh[E_EXPERTS];
  sh[e] = c;
  __syncthreads();
  if (e == 0) {
    int acc = 0;
    for (int i = 0; i < E_EXPERTS; ++i) { offsets[i] = acc; acc += sh[i]; }
  }
}

// ---------------------------------------------------------------------------
// Kernel 3: one wave (32 lanes) per expert: stable ballot-based prefix scan
// over tokens -> sorted_indices + permuted_probs (expert-major, token asc).
// ---------------------------------------------------------------------------
__global__ void scan_kernel(const unsigned long long* __restrict__ masks,
                            const int* __restrict__ offsets,
                            const float* __restrict__ token_probs,
                            int* __restrict__ sidx,
                            float* __restrict__ out_probs,
                            float* __restrict__ out_sidx) {
  int e    = blockIdx.x;    // 64 experts
  int lane = threadIdx.x;   // 32 lanes (wave32)
  int base = offsets[e];

  for (int t0 = 0; t0 < T_TOKENS; t0 += 32) {
    int t = t0 + lane;
    int bit = (int)((masks[t] >> e) & 1ull);
    unsigned int bal = (unsigned int)__ballot(bit);   // wave32: low 32 bits
    int prefix = __popc(bal & ((1u << lane) - 1u));
    if (bit) {
      int slot = base + prefix;
      sidx[slot]      = t;
      out_sidx[slot]  = (float)t;
      out_probs[slot] = token_probs[(size_t)t * E_EXPERTS + e];
    }
    base += __popc(bal);
  }
}

// ---------------------------------------------------------------------------
// Kernel 4: row gather. One wave per workgroup; each wave copies whole 8 KB
// rows through LDS with async B128 transfers (data never touches VGPRs).
// 16 async instrs x 512 B = 8 KB per row per phase. Loads RT (L2-resident
// source, 8x reuse); stores NT (write-once 512 MiB stream).
// ---------------------------------------------------------------------------
__global__ void gather_kernel(const float* __restrict__ hidden,
                              const int* __restrict__ sidx,
                              float* __restrict__ out) {
  int lane = threadIdx.x;   // 32
#if USE_ASYNC_LDS
  __shared__ __align__(16) char lbuf[ROW_BYTES];
#endif

  for (int row = (int)blockIdx.x; row < OUT_ROWS; row += (int)gridDim.x) {
    int src = sidx[row];

    // Prefetch the next row's source while this row is in flight.
    int nrow = row + (int)gridDim.x;
    if (nrow < OUT_ROWS) {
      int nsrc = sidx[nrow];
      __builtin_prefetch(hidden + (size_t)nsrc * H_DIM + lane * 4, 0, 0);
    }

    const char* s = (const char*)(hidden + (size_t)src * H_DIM);
    char*       d = (char*)(out + (size_t)row * H_DIM);

#if USE_ASYNC_LDS
#pragma unroll
    for (int k = 0; k < 16; ++k) {
      int off = k * 512 + lane * 16;
      __builtin_amdgcn_global_load_async_to_lds_b128(
          GPTR((void*)(s + off)), LPTR((void*)(lbuf + off)), 0, CPOL_RT);
    }
    wait_asynccnt0();   // loads landed in LDS
#pragma unroll
    for (int k = 0; k < 16; ++k) {
      int off = k * 512 + lane * 16;
      __builtin_amdgcn_global_store_async_from_lds_b128(
          GPTR((void*)(d + off)), LPTR((void*)(lbuf + off)), 0, CPOL_NT);
    }
    wait_asynccnt0();   // stores drained before LDS reuse
#else
    const float4* s4 = (const float4*)s;
    float4*       d4 = (float4*)d;
#pragma unroll
    for (int k = 0; k < 16; ++k) {
      d4[k * 32 + lane] = s4[k * 32 + lane];
    }
#endif
  }
}

// ---------------------------------------------------------------------------
// Launch wrapper
// ---------------------------------------------------------------------------
extern "C" void kernel_launch(void* const* d_in, const int* in_sizes, int n_in,
                              void* d_out, int out_size, void* d_ws, size_t ws_size,
                              hipStream_t stream) {
  const float* hidden = (const float*)d_in[0];   // (8192, 2048) f32
  const float* tprobs = (const float*)d_in[1];   // (8192, 64)   f32
  float* out = (float*)d_out;

  // Workspace layout
  char* ws = (char*)d_ws;
  unsigned long long* masks = (unsigned long long*)ws;           // 8192 * 8 B
  int* counts  = (int*)(ws + T_TOKENS * sizeof(unsigned long long));
  int* offsets = counts + E_EXPERTS;
  int* sidx    = offsets + E_EXPERTS;                            // 65536 ints

  // Output layout (flat, return order): tokens | counts | probs | indices
  float* out_tokens = out;
  float* out_counts = out + (size_t)OUT_ROWS * H_DIM;
  float* out_probs  = out_counts + E_EXPERTS;
  float* out_sidx   = out_probs + OUT_ROWS;

  topk_mask_kernel<<<T_TOKENS / 256, 256, 0, stream>>>(tprobs, masks);
  count_kernel<<<1, E_EXPERTS, 0, stream>>>(masks, counts, offsets, out_counts);
  scan_kernel<<<E_EXPERTS, 32, 0, stream>>>(masks, offsets, tprobs,
                                            sidx, out_probs, out_sidx);
  gather_kernel<<<8192, 32, 0, stream>>>(hidden, sidx, out_tokens);
}